// SetAbstraction_42975442764575
// MI455X (gfx1250) — compile-verified
//
#include <hip/hip_runtime.h>

// ---- problem constants ----
#define B_    8
#define N_    4096
#define DP    6
#define NPTS  1024
#define NSMP  32
#define RAD4  0.0016f      // (radius^2)^2 threshold for squared distance
#define C0    9
#define K0P   12           // K padded to multiple of 4
#define C1    64
#define C2    64
#define C3    128
#define MROWS (B_*NPTS*NSMP)   // 262144
#define BN_EPS 1e-3f

typedef float v2f __attribute__((ext_vector_type(2)));
typedef float v8f __attribute__((ext_vector_type(8)));

// ------------------------------------------------------------------
// 1) Farthest point sampling (reference quirk: distance to LAST point
//    only, masked; mask_new = min(dist*mask*1e11, mask)).
//    One block per batch; 4 points cached in registers per thread.
// ------------------------------------------------------------------
__global__ __launch_bounds__(1024) void fps_kernel(const float* __restrict__ xyz,
                                                   int* __restrict__ cidx) {
  const int b   = blockIdx.x;
  const int tid = threadIdx.x;
  const float* px = xyz + (size_t)b * N_ * 3;

  float X[4], Y[4], Z[4], MK[4];
#pragma unroll
  for (int j = 0; j < 4; ++j) {
    int i = tid + j * 1024;
    X[j] = px[i*3+0]; Y[j] = px[i*3+1]; Z[j] = px[i*3+2];
    MK[j] = 1.0f;
  }

  __shared__ float sWv[32];
  __shared__ int   sWi[32];
  __shared__ int   sBest;

  if (tid == 0) cidx[b*NPTS] = 0;
  int last = 0;

  for (int s = 0; s < NPTS - 1; ++s) {
    const float lx = px[last*3+0], ly = px[last*3+1], lz = px[last*3+2];
    float d[4];
    float bv = -1.0f; int bi = tid;
#pragma unroll
    for (int j = 0; j < 4; ++j) {
      int i = tid + j * 1024;
      float dx = X[j]-lx, dy = Y[j]-ly, dz = Z[j]-lz;
      float sq = dx*dx + dy*dy + dz*dz;
      float nrm = sq > 0.f ? sqrtf(sq) : 0.f;
      d[j] = nrm * MK[j];
      if (d[j] > bv) { bv = d[j]; bi = i; }   // strict > -> first index on ties
    }
    // wave32 reduction (argmax, ties -> smallest index)
    for (int off = 16; off > 0; off >>= 1) {
      float ov = __shfl_down(bv, off);
      int   oi = __shfl_down(bi, off);
      if (ov > bv || (ov == bv && oi < bi)) { bv = ov; bi = oi; }
    }
    const int wid = tid >> 5;
    if ((tid & 31) == 0) { sWv[wid] = bv; sWi[wid] = bi; }
    __syncthreads();
    if (wid == 0) {
      bv = sWv[tid & 31]; bi = sWi[tid & 31];
      for (int off = 16; off > 0; off >>= 1) {
        float ov = __shfl_down(bv, off);
        int   oi = __shfl_down(bi, off);
        if (ov > bv || (ov == bv && oi < bi)) { bv = ov; bi = oi; }
      }
      if (tid == 0) { sBest = bi; cidx[b*NPTS + s + 1] = bi; }
    }
    __syncthreads();
    last = sBest;
#pragma unroll
    for (int j = 0; j < 4; ++j)
      MK[j] = fminf(d[j] * MK[j] * 1e11f, MK[j]);
  }
}

// ------------------------------------------------------------------
// 2) cent_xyz = gather(xyz, cidx) -> d_out[0 .. 8*1024*3)
// ------------------------------------------------------------------
__global__ void centxyz_kernel(const float* __restrict__ xyz,
                               const int* __restrict__ cidx,
                               float* __restrict__ out) {
  int g = blockIdx.x * blockDim.x + threadIdx.x;
  if (g >= B_*NPTS) return;
  int b  = g / NPTS;
  int ci = cidx[g];
  const float* p = xyz + ((size_t)b * N_ + ci) * 3;
  out[g*3+0] = p[0]; out[g*3+1] = p[1]; out[g*3+2] = p[2];
}

// ------------------------------------------------------------------
// 3) query_ball: stable argsort of min(dist, r^2) == (sorted inside
//    points by (dist,idx)) followed by ascending-index outside fill.
//    One wave per centroid, 8 waves per block.
// ------------------------------------------------------------------
__global__ __launch_bounds__(256) void qball_kernel(const float* __restrict__ xyz,
                                                    const int* __restrict__ cidx,
                                                    int* __restrict__ gidx) {
  __shared__ unsigned long long skeys[8 * 64];
  __shared__ unsigned           scnt[8];

  const int w    = threadIdx.x >> 5;
  const int lane = threadIdx.x & 31;
  const int g    = blockIdx.x * 8 + w;       // centroid id, 0..8191
  const int b    = g >> 10;
  const float* px = xyz + (size_t)b * N_ * 3;

  if (lane == 0) scnt[w] = 0;
  __syncthreads();

  const int ci = cidx[g];
  const float cx = px[ci*3+0], cy = px[ci*3+1], cz = px[ci*3+2];

  for (int i = lane; i < N_; i += 32) {
    float dx = px[i*3+0]-cx, dy = px[i*3+1]-cy, dz = px[i*3+2]-cz;
    float sq = dx*dx + dy*dy + dz*dz;
    if (sq < RAD4) {
      float dn = sq > 0.f ? sqrtf(sq) : 0.f;
      unsigned pos = atomicAdd(&scnt[w], 1u);
      if (pos < 64) {
        unsigned long long key =
            ((unsigned long long)__float_as_uint(dn) << 32) | (unsigned)i;
        skeys[w*64 + pos] = key;
      }
    }
  }
  __syncthreads();

  unsigned m = scnt[w]; if (m > 64) m = 64;
  if (lane == 0) {               // tiny insertion sort (m is ~1 typically)
    unsigned long long* ks = &skeys[w*64];
    for (unsigned a = 1; a < m; ++a) {
      unsigned long long kv = ks[a];
      int q = (int)a - 1;
      while (q >= 0 && ks[q] > kv) { ks[q+1] = ks[q]; --q; }
      ks[q+1] = kv;
    }
  }
  __syncthreads();

  unsigned nIn = m < 32u ? m : 32u;
  if (lane < (int)nIn)
    gidx[g*NSMP + lane] = (int)(unsigned)(skeys[w*64 + lane] & 0xffffffffu);
  if (lane == 0 && nIn < 32u) {  // fill with smallest-index outside points
    int slot = (int)nIn;
    for (int i = 0; i < N_ && slot < 32; ++i) {
      float dx = px[i*3+0]-cx, dy = px[i*3+1]-cy, dz = px[i*3+2]-cz;
      float sq = dx*dx + dy*dy + dz*dz;
      if (!(sq < RAD4)) { gidx[g*NSMP + slot] = i; ++slot; }
    }
  }
}

// ------------------------------------------------------------------
// 4) Build X0 [MROWS, 12]: concat(group_xyz, group_pts, zero pad)
// ------------------------------------------------------------------
__global__ void build_x0(const float* __restrict__ xyz,
                         const float* __restrict__ pts,
                         const int* __restrict__ gidx,
                         float* __restrict__ X0) {
  int r = blockIdx.x * blockDim.x + threadIdx.x;
  if (r >= MROWS) return;
  int b   = r >> 15;               // / (1024*32)
  int idx = gidx[r];
  const float* p3 = xyz + ((size_t)b * N_ + idx) * 3;
  const float* p6 = pts + ((size_t)b * N_ + idx) * DP;
  float* o = X0 + (size_t)r * K0P;
  o[0]=p3[0]; o[1]=p3[1]; o[2]=p3[2];
  o[3]=p6[0]; o[4]=p6[1]; o[5]=p6[2]; o[6]=p6[3]; o[7]=p6[4]; o[8]=p6[5];
  o[9]=0.f; o[10]=0.f; o[11]=0.f;
}

__global__ void pad_w0(const float* __restrict__ w0, float* __restrict__ w0p) {
  int i = blockIdx.x * blockDim.x + threadIdx.x;
  if (i >= K0P * C1) return;
  int k = i / C1, n = i % C1;
  w0p[i] = (k < C0) ? w0[k*C1 + n] : 0.f;
}

// ------------------------------------------------------------------
// 5) FP32 WMMA GEMM, 64x16 tile per wave (4 M-subtiles reuse each B
//    fragment). FUSED=true applies the previous layer's BN+ReLU on
//    the A-load: a' = max(0, a*scale[k] + shift[k]) with
//    ss = [scale[K], shift[K]]. Compile-time flag -> branch-free loop.
//    A layout: lanes 0-15 hold (K0,K1) of row M=lane; lanes 16-31 (K2,K3).
//    D layout: vD[r] -> row = r + 8*(lane>=16), col = lane&15.
// ------------------------------------------------------------------
template <bool FUSED>
__global__ __launch_bounds__(32) void gemm_wmma_f32(const float* __restrict__ Xm,
                                                    const float* __restrict__ Wm,
                                                    float* __restrict__ Ym,
                                                    const float* __restrict__ ss,
                                                    int N, int K) {
  const int lane = threadIdx.x;
  const int half = lane >> 4;
  const int l16  = lane & 15;
  const int row  = blockIdx.x * 64;
  const int col  = blockIdx.y * 16;
  const float* xr = Xm + (size_t)(row + l16) * K;

  v8f acc[4];
#pragma unroll
  for (int t = 0; t < 4; ++t) acc[t] = (v8f){0.f,0.f,0.f,0.f,0.f,0.f,0.f,0.f};

  for (int kk = 0; kk < K; kk += 4) {
    const int k0 = kk + 2*half;
    v2f bfrag;
    bfrag.x = Wm[(size_t)(k0 + 0) * N + col + l16];
    bfrag.y = Wm[(size_t)(k0 + 1) * N + col + l16];
    float sc0, sh0, sc1, sh1;
    if (FUSED) { sc0 = ss[k0]; sh0 = ss[K + k0]; sc1 = ss[k0+1]; sh1 = ss[K + k0 + 1]; }
#pragma unroll
    for (int t = 0; t < 4; ++t) {
      const float* xt = xr + (size_t)(t * 16) * K;
      v2f a;
      a.x = xt[k0];
      a.y = xt[k0 + 1];
      if (FUSED) {
        a.x = fmaxf(0.f, a.x * sc0 + sh0);
        a.y = fmaxf(0.f, a.y * sc1 + sh1);
      }
      acc[t] = __builtin_amdgcn_wmma_f32_16x16x4_f32(false, a, false, bfrag,
                                                     (short)0, acc[t], false, false);
    }
  }
#pragma unroll
  for (int t = 0; t < 4; ++t) {
    const int r0 = row + t * 16 + half * 8;
#pragma unroll
    for (int r = 0; r < 8; ++r)
      Ym[(size_t)(r0 + r) * N + col + l16] = acc[t][r];
  }
}

// ------------------------------------------------------------------
// 6) BatchNorm stats on RAW GEMM output (deterministic two-stage);
//    final stage emits per-channel scale/shift for fused consumers.
//    Dense bias is skipped: it cancels exactly in (h - mean).
// ------------------------------------------------------------------
__global__ void bn_part(const float* __restrict__ Y, float* __restrict__ part, int C) {
  int c = threadIdx.x, blk = blockIdx.x;
  const int rows = MROWS / 512;
  const float* p = Y + (size_t)blk * rows * C + c;
  float s = 0.f, s2 = 0.f;
  for (int r = 0; r < rows; ++r) { float v = p[(size_t)r * C]; s += v; s2 += v*v; }
  part[blk*C + c]         = s;
  part[512*C + blk*C + c] = s2;
}

__global__ void bn_final(const float* __restrict__ part, float* __restrict__ ss,
                         const float* __restrict__ g, const float* __restrict__ be, int C) {
  int c = threadIdx.x;
  float s = 0.f, s2 = 0.f;
  for (int b = 0; b < 512; ++b) { s += part[b*C + c]; s2 += part[512*C + b*C + c]; }
  float mean = s / (float)MROWS;
  float var  = s2 / (float)MROWS - mean * mean;
  float sc = g[c] * rsqrtf(var + BN_EPS);
  ss[c]     = sc;
  ss[C + c] = be[c] - mean * sc;       // y = raw*sc + shift, then relu
}

// ------------------------------------------------------------------
// 7) Fused BN+ReLU + max-pool over the 32 samples -> d_out[24576 ..)
// ------------------------------------------------------------------
__global__ void maxpool_kernel(const float* __restrict__ Y3,
                               const float* __restrict__ ss,
                               float* __restrict__ out) {
  int i = blockIdx.x * blockDim.x + threadIdx.x;
  if (i >= B_*NPTS*C3) return;
  int rc = i / C3, ch = i % C3;
  const float sc = ss[ch], sh = ss[C3 + ch];
  const float* p = Y3 + (size_t)rc * NSMP * C3 + ch;
  float m = fmaxf(0.f, p[0] * sc + sh);
  for (int s = 1; s < NSMP; ++s)
    m = fmaxf(m, fmaxf(0.f, p[(size_t)s * C3] * sc + sh));
  out[i] = m;
}

// ------------------------------------------------------------------
extern "C" void kernel_launch(void* const* d_in, const int* in_sizes, int n_in,
                              void* d_out, int out_size, void* d_ws, size_t ws_size,
                              hipStream_t stream) {
  const float* xyz = (const float*)d_in[0];
  const float* pts = (const float*)d_in[1];
  const float* w0  = (const float*)d_in[2];
  const float* g0  = (const float*)d_in[4];
  const float* be0 = (const float*)d_in[5];
  const float* w1  = (const float*)d_in[6];
  const float* g1  = (const float*)d_in[8];
  const float* be1 = (const float*)d_in[9];
  const float* w2  = (const float*)d_in[10];
  const float* g2  = (const float*)d_in[12];
  const float* be2 = (const float*)d_in[13];
  float* out = (float*)d_out;

  char* ws = (char*)d_ws;
  size_t off = 0;
  auto alloc = [&](size_t bytes) -> void* {
    void* p = ws + off;
    off = (off + bytes + 255) & ~(size_t)255;
    return p;
  };
  int*   cidx = (int*)  alloc((size_t)B_*NPTS*4);
  int*   gidx = (int*)  alloc((size_t)B_*NPTS*NSMP*4);
  float* w0p  = (float*)alloc((size_t)K0P*C1*4);
  float* part = (float*)alloc((size_t)2*512*C3*4);
  float* ss1  = (float*)alloc((size_t)2*C1*4);
  float* ss2  = (float*)alloc((size_t)2*C2*4);
  float* ss3  = (float*)alloc((size_t)2*C3*4);
  float* X0   = (float*)alloc((size_t)MROWS*K0P*4);
  float* Y1   = (float*)alloc((size_t)MROWS*C1*4);
  float* Y2   = (float*)alloc((size_t)MROWS*C2*4);
  float* Y3   = (float*)alloc((size_t)MROWS*C3*4);

  // FPS + centroid gather
  fps_kernel<<<B_, 1024, 0, stream>>>(xyz, cidx);
  centxyz_kernel<<<(B_*NPTS + 255)/256, 256, 0, stream>>>(xyz, cidx, out);

  // ball query + grouping
  qball_kernel<<<(B_*NPTS)/8, 256, 0, stream>>>(xyz, cidx, gidx);
  build_x0<<<(MROWS + 255)/256, 256, 0, stream>>>(xyz, pts, gidx, X0);
  pad_w0<<<(K0P*C1 + 255)/256, 256, 0, stream>>>(w0, w0p);

  // layer 1: 12 -> 64  (raw input)
  gemm_wmma_f32<false><<<dim3(MROWS/64, C1/16), 32, 0, stream>>>(X0, w0p, Y1, nullptr, C1, K0P);
  bn_part <<<512, C1, 0, stream>>>(Y1, part, C1);
  bn_final<<<1,   C1, 0, stream>>>(part, ss1, g0, be0, C1);

  // layer 2: 64 -> 64  (BN1+ReLU fused into A-load)
  gemm_wmma_f32<true><<<dim3(MROWS/64, C2/16), 32, 0, stream>>>(Y1, w1, Y2, ss1, C2, C1);
  bn_part <<<512, C2, 0, stream>>>(Y2, part, C2);
  bn_final<<<1,   C2, 0, stream>>>(part, ss2, g1, be1, C2);

  // layer 3: 64 -> 128 (BN2+ReLU fused into A-load)
  gemm_wmma_f32<true><<<dim3(MROWS/64, C3/16), 32, 0, stream>>>(Y2, w2, Y3, ss2, C3, C2);
  bn_part <<<512, C3, 0, stream>>>(Y3, part, C3);
  bn_final<<<1,   C3, 0, stream>>>(part, ss3, g2, be2, C3);

  // fused BN3+ReLU + max-pool over samples
  maxpool_kernel<<<(B_*NPTS*C3 + 255)/256, 256, 0, stream>>>(Y3, ss3, out + (size_t)B_*NPTS*3);
}